// ScaledDotProductAttention_52707838656800
// MI455X (gfx1250) — compile-verified
//
#include <hip/hip_runtime.h>

typedef __attribute__((ext_vector_type(2))) float v2f;
typedef __attribute__((ext_vector_type(4))) float v4f;
typedef __attribute__((ext_vector_type(8))) float v8f;

#define B_ 4
#define H_ 16
#define S_ 2048
#define D_ 64
#define LDSTR 2052   // padded LDS row stride in floats: 16B-aligned, bank-spread

__launch_bounds__(128)
__global__ void sdpa_f32_wmma_kernel(const float* __restrict__ Q,
                                     const float* __restrict__ K,
                                     const float* __restrict__ V,
                                     const unsigned char* __restrict__ mask,
                                     float* __restrict__ ctx,
                                     float* __restrict__ attn)
{
    __shared__ float sc[16 * LDSTR];   // 16 query rows x 2048 scores (padded)
    __shared__ float red[128];
    __shared__ float rowmax[16];
    __shared__ float rowinv[16];

    const int tid  = threadIdx.x;
    const int w    = tid >> 5;          // wave 0..3
    const int lane = tid & 31;
    const int half = lane >> 4;         // 0 or 1 (lane group)
    const int l16  = lane & 15;

    const int qt = blockIdx.x;          // 128 query tiles of 16
    const int h  = blockIdx.y;
    const int b  = blockIdx.z;
    const int qbase = qt * 16;

    const size_t bh = (size_t)b * H_ + h;
    const float* Qb = Q + (bh * S_ + qbase) * D_;
    const float* Kb = K + bh * (size_t)S_ * D_;
    const float* Vb = V + bh * (size_t)S_ * D_;
    const unsigned char* Mb = mask + (size_t)b * S_;   // [B,1,1,S] key-padding mask
    float* Cb = ctx  + (bh * S_ + qbase) * D_;
    float* Ab = attn + (bh * S_ + qbase) * (size_t)S_;

    // ---------- Phase 1: scores = scale * Q K^T, masked, into LDS ----------
    // A-fragment (16x4 f32 per WMMA): lane holds Q[l16][4c + 2*half + {0,1}]
    v2f a[16];
    #pragma unroll
    for (int c = 0; c < 16; ++c)
        a[c] = *(const v2f*)(Qb + l16 * D_ + c * 4 + half * 2);

    const float scale = 0.125f;   // 1/sqrt(64)

    for (int kt = w; kt < S_ / 16; kt += 4) {
        // B-fragment: B[d][col] = K[kt*16+col][d]; lane col = l16
        const float* Kt = Kb + (size_t)(kt * 16 + l16) * D_ + half * 2;
        v8f acc = {};
        #pragma unroll
        for (int c = 0; c < 16; ++c) {
            v2f bf = *(const v2f*)(Kt + c * 4);
            acc = __builtin_amdgcn_wmma_f32_16x16x4_f32(
                      false, a[c], false, bf, (short)0, acc, false, false);
        }
        const int kcol = kt * 16 + l16;
        const bool msk = (Mb[kcol] != 0);
        #pragma unroll
        for (int r = 0; r < 8; ++r) {
            int m = r + half * 8;                    // C/D layout: M = r + 8*half
            sc[m * LDSTR + kcol] = msk ? -1e9f : acc[r] * scale;
        }
    }
    __syncthreads();

    // ---------- Phase 2: row max, then e=exp(s-m) stored in place + sum ----
    {
        const int row = tid & 15;
        const int seg = tid >> 4;                    // 8 segments x 256 cols
        float* rowp = &sc[row * LDSTR + seg * 256];

        float mloc = -3.402823466e38f;
        for (int j = 0; j < 64; ++j) {               // vectorized b128 scans
            v4f s4 = *(const v4f*)(rowp + j * 4);
            mloc = fmaxf(mloc, fmaxf(fmaxf(s4.x, s4.y), fmaxf(s4.z, s4.w)));
        }
        red[tid] = mloc;
        __syncthreads();
        if (tid < 16) {
            float m = red[tid];
            for (int s2 = 1; s2 < 8; ++s2) m = fmaxf(m, red[tid + 16 * s2]);
            rowmax[tid] = m;
        }
        __syncthreads();

        const float m = rowmax[row];
        float sloc = 0.f;
        for (int j = 0; j < 64; ++j) {
            v4f s4 = *(const v4f*)(rowp + j * 4);
            v4f e4;
            e4.x = __expf(s4.x - m);
            e4.y = __expf(s4.y - m);
            e4.z = __expf(s4.z - m);
            e4.w = __expf(s4.w - m);
            *(v4f*)(rowp + j * 4) = e4;              // keep e for normalize + PV
            sloc += (e4.x + e4.y) + (e4.z + e4.w);
        }
        red[tid] = sloc;
        __syncthreads();
        if (tid < 16) {
            float ssum = red[tid];
            for (int s2 = 1; s2 < 8; ++s2) ssum += red[tid + 16 * s2];
            rowinv[tid] = 1.0f / ssum;
        }
        __syncthreads();
    }

    // ---------- Phase 3: p = e * rinv -> LDS + coalesced attn writes -------
    for (int row = 0; row < 16; ++row) {
        const float rinv = rowinv[row];
        float* rl = &sc[row * LDSTR];
        float* rg = Ab + (size_t)row * S_;
        for (int c4 = tid; c4 < S_ / 4; c4 += 128) {
            const int c = c4 * 4;
            v4f e4 = *(const v4f*)(rl + c);
            v4f p4;
            p4.x = e4.x * rinv;
            p4.y = e4.y * rinv;
            p4.z = e4.z * rinv;
            p4.w = e4.w * rinv;
            *(v4f*)(rl + c) = p4;        // probabilities for PV phase
            *(v4f*)(rg + c) = p4;        // stream attn to HBM (coalesced b128)
        }
    }
    __syncthreads();

    // ---------- Phase 4: context = P V; wave w owns D columns [16w,16w+16) --
    // Two independent accumulator chains (even/odd chunks) for WMMA ILP.
    {
        const int dcol = w * 16 + l16;
        v8f cacc0 = {};
        v8f cacc1 = {};
        for (int chunk = 0; chunk < S_ / 4; chunk += 2) {
            const int kb0 = chunk * 4 + half * 2;
            const int kb1 = kb0 + 4;
            v2f pa0 = *(const v2f*)(&sc[l16 * LDSTR + kb0]);
            v2f pa1 = *(const v2f*)(&sc[l16 * LDSTR + kb1]);
            v2f bb0, bb1;
            bb0.x = Vb[(size_t)(kb0 + 0) * D_ + dcol];
            bb0.y = Vb[(size_t)(kb0 + 1) * D_ + dcol];
            bb1.x = Vb[(size_t)(kb1 + 0) * D_ + dcol];
            bb1.y = Vb[(size_t)(kb1 + 1) * D_ + dcol];
            cacc0 = __builtin_amdgcn_wmma_f32_16x16x4_f32(
                        false, pa0, false, bb0, (short)0, cacc0, false, false);
            cacc1 = __builtin_amdgcn_wmma_f32_16x16x4_f32(
                        false, pa1, false, bb1, (short)0, cacc1, false, false);
        }
        #pragma unroll
        for (int r = 0; r < 8; ++r) {
            int m = r + half * 8;
            Cb[(size_t)m * D_ + dcol] = cacc0[r] + cacc1[r];
        }
    }
}

extern "C" void kernel_launch(void* const* d_in, const int* in_sizes, int n_in,
                              void* d_out, int out_size, void* d_ws, size_t ws_size,
                              hipStream_t stream) {
    (void)in_sizes; (void)n_in; (void)out_size; (void)d_ws; (void)ws_size;
    const float* Q = (const float*)d_in[0];
    const float* K = (const float*)d_in[1];
    const float* V = (const float*)d_in[2];
    const unsigned char* mask = (const unsigned char*)d_in[3];  // jnp.bool_ = 1 byte

    float* ctx  = (float*)d_out;                                   // [B,H,S,D] first
    float* attn = (float*)d_out + (size_t)B_ * H_ * S_ * D_;       // then [B,H,S,S]

    dim3 grid(S_ / 16, H_, B_);   // 128 x 16 x 4 = 8192 workgroups
    sdpa_f32_wmma_kernel<<<grid, 128, 0, stream>>>(Q, K, V, mask, ctx, attn);
}